// VNGNN_46188078301271
// MI455X (gfx1250) — compile-verified
//
#include <hip/hip_runtime.h>

// ---------------- problem constants (match reference) ----------------
#define NN   50000
#define EE   800000
#define FINC 128
#define HC   256
#define OUTC 128
#define EPSF 1e-5f

typedef float v2f __attribute__((ext_vector_type(2)));
typedef float v8f __attribute__((ext_vector_type(8)));

// ---------------- small utility kernels ----------------
__global__ void k_fill(float* __restrict__ p, int n, float v) {
    int i = blockIdx.x * blockDim.x + threadIdx.x;
    if (i < n) p[i] = v;
}

__global__ void k_deg_accum(const int* __restrict__ ei, float* __restrict__ deg) {
    int e = blockIdx.x * blockDim.x + threadIdx.x;
    if (e < EE) atomicAdd(&deg[ei[EE + e]], 1.0f);
}

__global__ void k_deg_inv(float* __restrict__ deg) {
    int i = blockIdx.x * blockDim.x + threadIdx.x;
    if (i < NN) deg[i] = rsqrtf(deg[i]);   // deg -> d^{-1/2} in place
}

__global__ void k_init_vn(const float* __restrict__ vnp, float* __restrict__ vn) {
    int i = threadIdx.x;                   // block of 256
    vn[i] = (i < FINC) ? vnp[i] : 0.0f;
}

// W[k][n] (din x dout, row-major)  ->  Wt[n][k] (dout x din)
__global__ void k_transpose_w(const float* __restrict__ W, float* __restrict__ Wt,
                              int din, int dout) {
    int i = blockIdx.x * blockDim.x + threadIdx.x;
    if (i >= din * dout) return;
    int n = i % dout;
    int k = i / dout;                      // coalesced read along n
    Wt[(size_t)n * din + k] = W[i];
}

// ---------------- GEMM: z = (h + vn) @ W via V_WMMA_F32_16X16X4_F32 --------
// One wave computes a 16x32 tile of z (two accumulators; A fragment reused).
// blockDim.x = dout (4 waves for dout=128, 8 waves for dout=256);
// wave w owns N-columns [32w, 32w+32). grid.x = NN/16 M-tiles.
__global__ void k_gemm_wmma(const float* __restrict__ h,
                            const float* __restrict__ Wt,   // dout x din
                            const float* __restrict__ vn,
                            float* __restrict__ z,
                            int din, int dout) {
    __shared__ float vns[HC];
    for (int i = threadIdx.x; i < din; i += blockDim.x) vns[i] = vn[i];
    __syncthreads();

    const int waveId = threadIdx.x >> 5;
    const int lane   = threadIdx.x & 31;
    const int half   = lane >> 4;          // 0: lanes 0-15 (K=k0,k0+1), 1: lanes 16-31 (K=k0+2,k0+3)
    const int lm     = lane & 15;

    const int m0 = blockIdx.x * 16;
    const int n0 = waveId * 32;

    v8f c0 = {}, c1 = {};
    const float* hrow = h  + (size_t)(m0 + lm) * din;        // A row m0+lm
    const float* w0   = Wt + (size_t)(n0 + lm) * din;        // B column n0+lm
    const float* w1   = Wt + (size_t)(n0 + 16 + lm) * din;   // B column n0+16+lm

    #pragma unroll 4
    for (int k0 = 0; k0 < din; k0 += 4) {
        const int kb = k0 + 2 * half;
        v2f a;  a.x  = hrow[kb] + vns[kb];  a.y  = hrow[kb + 1] + vns[kb + 1];
        v2f b0; b0.x = w0[kb];              b0.y = w0[kb + 1];
        v2f b1; b1.x = w1[kb];              b1.y = w1[kb + 1];
        c0 = __builtin_amdgcn_wmma_f32_16x16x4_f32(false, a, false, b0,
                                                   (short)0, c0, false, false);
        c1 = __builtin_amdgcn_wmma_f32_16x16x4_f32(false, a, false, b1,
                                                   (short)0, c1, false, false);
    }
    // C/D layout: VGPR v -> row (v + 8*half), col = lane&15
    for (int v = 0; v < 8; ++v) {
        const size_t row = (size_t)(m0 + v + 8 * half) * dout;
        z[row + n0 + lm]      = c0[v];
        z[row + n0 + 16 + lm] = c1[v];
    }
}

// ---------------- agg init: self-loop + bias ----------------
__global__ void k_init_agg(const float* __restrict__ z,
                           const float* __restrict__ dinv,
                           const float* __restrict__ bias,
                           float* __restrict__ agg,
                           int dout, int dshift) {
    int i = blockIdx.x * blockDim.x + threadIdx.x;
    if (i >= NN * dout) return;
    int c = i & (dout - 1);
    int n = i >> dshift;
    float di = dinv[n];
    agg[i] = z[i] * di * di + bias[c];
}

// ---------------- edge scatter: one wave32 per edge ----------------
__global__ void k_scatter(const float* __restrict__ z,
                          const int* __restrict__ ei,
                          const float* __restrict__ dinv,
                          float* __restrict__ agg,
                          int dout) {
    int wave = (blockIdx.x * blockDim.x + threadIdx.x) >> 5;
    int lane = threadIdx.x & 31;
    if (wave >= EE) return;
    int s = ei[wave];
    int d = ei[EE + wave];
    float w = dinv[s] * dinv[d];
    const float* zr = z + (size_t)s * dout;
    float* ar = agg + (size_t)d * dout;
    for (int c = lane; c < dout; c += 32)
        atomicAdd(&ar[c], zr[c] * w);
}

// ---------------- column statistics (sum / sumsq), blockDim = cols ----------
__global__ void k_colstats(const float* __restrict__ src,
                           float* __restrict__ colsum,
                           float* __restrict__ colsq,   // may be nullptr
                           int cols, int rowsPerBlock) {
    int c  = threadIdx.x;
    int r0 = blockIdx.x * rowsPerBlock;
    int r1 = r0 + rowsPerBlock; if (r1 > NN) r1 = NN;
    float s = 0.f, s2 = 0.f;
    for (int r = r0; r < r1; ++r) {
        float v = src[(size_t)r * cols + c];
        s += v; s2 += v * v;
    }
    atomicAdd(&colsum[c], s);
    if (colsq) atomicAdd(&colsq[c], s2);
}

__global__ void k_bn_finalize(const float* __restrict__ colsum,
                              const float* __restrict__ colsq,
                              const float* __restrict__ g,
                              const float* __restrict__ b,
                              float* __restrict__ bn_a,
                              float* __restrict__ bn_sh) {
    int c = threadIdx.x;
    float mean = colsum[c] * (1.0f / NN);
    float var  = colsq[c] * (1.0f / NN) - mean * mean;     // biased, as jnp.var
    float a    = g[c] * rsqrtf(var + EPSF);
    bn_a[c]  = a;
    bn_sh[c] = b[c] - mean * a;
}

__global__ void k_bn_apply(const float* __restrict__ agg,
                           const float* __restrict__ bn_a,
                           const float* __restrict__ bn_sh,
                           float* __restrict__ out,
                           int dout, int dshift) {
    int i = blockIdx.x * blockDim.x + threadIdx.x;
    if (i >= NN * dout) return;
    int c = i & (dout - 1);
    out[i] = fmaxf(0.0f, agg[i] * bn_a[c] + bn_sh[c]);
    (void)dshift;
}

__global__ void k_pool_finalize(const float* __restrict__ colsum,
                                const float* __restrict__ vn,
                                float* __restrict__ pooled,
                                int din) {
    int t = threadIdx.x;                 // block of 256
    if (t < din) pooled[t] = colsum[t] + vn[t];
}

// ---------------- virtual-node MLP: one 512-thread block ----------------
__global__ void k_vn_mlp(const float* __restrict__ pooled,
                         const float* __restrict__ w1, const float* __restrict__ b1,
                         const float* __restrict__ lg1, const float* __restrict__ lb1,
                         const float* __restrict__ w2, const float* __restrict__ b2,
                         const float* __restrict__ lg2, const float* __restrict__ lb2,
                         float* __restrict__ vn_out, int din) {
    const int DMID = 512, DOUT = 256;
    __shared__ float sp[HC];
    __shared__ float h1[512];
    __shared__ float red[512];
    int t = threadIdx.x;
    if (t < din) sp[t] = pooled[t];
    __syncthreads();

    // layer 1: din -> 512, ReLU
    float acc = b1[t];
    for (int i = 0; i < din; ++i) acc = fmaf(sp[i], w1[(size_t)i * DMID + t], acc);
    acc = fmaxf(acc, 0.0f);

    // LayerNorm over 512
    red[t] = acc; __syncthreads();
    for (int s = 256; s > 0; s >>= 1) { if (t < s) red[t] += red[t + s]; __syncthreads(); }
    float mean = red[0] * (1.0f / DMID); __syncthreads();
    float d = acc - mean;
    red[t] = d * d; __syncthreads();
    for (int s = 256; s > 0; s >>= 1) { if (t < s) red[t] += red[t + s]; __syncthreads(); }
    float var = red[0] * (1.0f / DMID); __syncthreads();
    h1[t] = d * rsqrtf(var + EPSF) * lg1[t] + lb1[t];
    __syncthreads();

    // layer 2: 512 -> 256, ReLU
    float acc2 = 0.0f;
    if (t < DOUT) {
        acc2 = b2[t];
        for (int i = 0; i < DMID; ++i) acc2 = fmaf(h1[i], w2[(size_t)i * DOUT + t], acc2);
        acc2 = fmaxf(acc2, 0.0f);
    }
    // LayerNorm over 256
    red[t] = (t < DOUT) ? acc2 : 0.0f; __syncthreads();
    for (int s = 256; s > 0; s >>= 1) { if (t < s) red[t] += red[t + s]; __syncthreads(); }
    float mean2 = red[0] * (1.0f / DOUT); __syncthreads();
    float d2 = acc2 - mean2;
    red[t] = (t < DOUT) ? d2 * d2 : 0.0f; __syncthreads();
    for (int s = 256; s > 0; s >>= 1) { if (t < s) red[t] += red[t + s]; __syncthreads(); }
    float var2 = red[0] * (1.0f / DOUT); __syncthreads();
    if (t < DOUT) vn_out[t] = d2 * rsqrtf(var2 + EPSF) * lg2[t] + lb2[t];
}

// ---------------- host orchestration ----------------
extern "C" void kernel_launch(void* const* d_in, const int* in_sizes, int n_in,
                              void* d_out, int out_size, void* d_ws, size_t ws_size,
                              hipStream_t stream) {
    (void)in_sizes; (void)n_in; (void)out_size; (void)ws_size;
    // Input order (setup_inputs dict insertion order, params flattened
    // conv_w[0..2], conv_b[0..2], bn_g[0..2], bn_b[0..2], vn, mlp[0]{w1,b1,ln1_g,
    // ln1_b,w2,b2,ln2_g,ln2_b}, mlp[1]{...}):
    const float* x  = (const float*)d_in[0];
    const int*   ei = (const int*)d_in[1];
    const float* Wc[3], *Bc[3], *Gn[3], *Bn[3];
    for (int l = 0; l < 3; ++l) {
        Wc[l] = (const float*)d_in[2 + l];
        Bc[l] = (const float*)d_in[5 + l];
        Gn[l] = (const float*)d_in[8 + l];
        Bn[l] = (const float*)d_in[11 + l];
    }
    const float* vnp = (const float*)d_in[14];
    const float* mlp[2][8];
    for (int j = 0; j < 8; ++j) { mlp[0][j] = (const float*)d_in[15 + j];
                                  mlp[1][j] = (const float*)d_in[23 + j]; }

    // workspace layout
    float* w = (float*)d_ws;
    const size_t NH = (size_t)NN * HC;
    float* dinv   = w;                 // NN (deg -> d^{-1/2})
    float* Z      = w + 50048;         // NN*HC
    float* bufA   = Z + NH;            // NN*HC
    float* bufB   = bufA + NH;         // NN*HC
    float* colsum = bufB + NH;         // 256
    float* colsq  = colsum + 256;      // 256
    float* bn_a   = colsq + 256;       // 256
    float* bn_sh  = bn_a + 256;        // 256
    float* vn     = bn_sh + 256;       // 256
    float* pooled = vn + 256;          // 256
    float* Wt     = pooled + 256;      // up to 256*256 (transposed layer weight)

    // degrees + normalization
    k_fill<<<(NN + 255) / 256, 256, 0, stream>>>(dinv, NN, 1.0f);
    k_deg_accum<<<(EE + 255) / 256, 256, 0, stream>>>(ei, dinv);
    k_deg_inv<<<(NN + 255) / 256, 256, 0, stream>>>(dinv);
    k_init_vn<<<1, 256, 0, stream>>>(vnp, vn);

    const int dins[3]  = {FINC, HC, HC};
    const int douts[3] = {HC, HC, OUTC};
    const float* hin[3] = {x, bufA, bufB};
    float* aggb[3] = {bufA, bufB, bufA};
    float* outh[3] = {bufA, bufB, (float*)d_out};

    const int ROWS = 128;  // rows per block for column-stat kernels

    for (int l = 0; l < 3; ++l) {
        const int din = dins[l], dout = douts[l];
        const int dsh = (dout == 256) ? 8 : 7;
        const int tot = NN * dout;

        // W -> Wt (dout x din), then z = (h + vn) @ W via fp32 WMMA
        k_transpose_w<<<(din * dout + 255) / 256, 256, 0, stream>>>(Wc[l], Wt, din, dout);
        k_gemm_wmma<<<NN / 16, dout, 0, stream>>>(hin[l], Wt, vn, Z, din, dout);

        // agg = z*selfw + bias, then scatter edges
        k_init_agg<<<(tot + 255) / 256, 256, 0, stream>>>(Z, dinv, Bc[l], aggb[l], dout, dsh);
        k_scatter<<<EE / 8, 256, 0, stream>>>(Z, ei, dinv, aggb[l], dout);

        // BatchNorm (training stats) + ReLU
        k_fill<<<2, 256, 0, stream>>>(colsum, 512, 0.0f);   // zero colsum+colsq
        k_colstats<<<(NN + ROWS - 1) / ROWS, dout, 0, stream>>>(aggb[l], colsum, colsq, dout, ROWS);
        k_bn_finalize<<<1, dout, 0, stream>>>(colsum, colsq, Gn[l], Bn[l], bn_a, bn_sh);
        k_bn_apply<<<(tot + 255) / 256, 256, 0, stream>>>(aggb[l], bn_a, bn_sh, outh[l], dout, dsh);

        // virtual-node update (uses pre-conv h)
        if (l < 2) {
            k_fill<<<1, 256, 0, stream>>>(colsum, 256, 0.0f);
            k_colstats<<<(NN + ROWS - 1) / ROWS, din, 0, stream>>>(hin[l], colsum, nullptr, din, ROWS);
            k_pool_finalize<<<1, 256, 0, stream>>>(colsum, vn, pooled, din);
            k_vn_mlp<<<1, 512, 0, stream>>>(pooled,
                                            mlp[l][0], mlp[l][1], mlp[l][2], mlp[l][3],
                                            mlp[l][4], mlp[l][5], mlp[l][6], mlp[l][7],
                                            vn, din);
        }
    }
}